// Model_32796370272774
// MI455X (gfx1250) — compile-verified
//
#include <hip/hip_runtime.h>
#include <hip/hip_bf16.h>
#include <math.h>

typedef __attribute__((ext_vector_type(2))) float v2f;
typedef __attribute__((ext_vector_type(8))) float v8f;

#define L_DIM  2048
#define E_DIM  512
#define EPT    300
#define EPTP   304   // padded to multiple of 4 for K-steps
#define NHEAD  16
#define NEGINF 1e20f
#define THRESH 0.1f

// ---------------------------------------------------------------------------
// CDNA5 async global->LDS copy (ASYNCcnt-tracked). SADDR form:
//   global_load_async_to_lds_b128 vdst(lds byte addr), vaddr(u32 byte off), saddr
// ---------------------------------------------------------------------------
__device__ __forceinline__ void async_g2l_b128(unsigned lds_byte_addr,
                                               unsigned byte_off,
                                               const float* base) {
    asm volatile("global_load_async_to_lds_b128 %0, %1, %2"
                 :
                 : "v"(lds_byte_addr), "v"(byte_off), "s"(base)
                 : "memory");
}

__device__ __forceinline__ void wait_async0() {
#if __has_builtin(__builtin_amdgcn_s_wait_asynccnt)
    __builtin_amdgcn_s_wait_asynccnt(0);
#else
    asm volatile("s_wait_asynccnt 0x0" ::: "memory");
#endif
}

__device__ __forceinline__ unsigned lds_addr_of(const void* p) {
    return (unsigned)(size_t)p;   // generic LDS ptr: low 32 bits = DS address
}

// ---------------------------------------------------------------------------
// Prep 1: w2[h][e] = W_pt[h][e]^2, zero-padded to EPTP columns.
// ---------------------------------------------------------------------------
__global__ void w2_kernel(const float* __restrict__ Wpt, float* __restrict__ w2) {
    int i = blockIdx.x * blockDim.x + threadIdx.x;
    if (i >= NHEAD * EPTP) return;
    int h = i / EPTP, e = i % EPTP;
    float w = (e < EPT) ? Wpt[h * EPT + e] : 0.0f;
    w2[i] = w * w;
}

// ---------------------------------------------------------------------------
// Prep 2: zero-padded PT copy [L_DIM x EPTP] -> branch-free WMMA K-loops.
// ---------------------------------------------------------------------------
__global__ void pad_pt_kernel(const float* __restrict__ PT, float* __restrict__ ptp) {
    int i = blockIdx.x * blockDim.x + threadIdx.x;
    if (i >= L_DIM * EPTP) return;
    int l = i / EPTP, e = i % EPTP;
    ptp[i] = (e < EPT) ? PT[l * EPT + e] : 0.0f;
}

// ---------------------------------------------------------------------------
// Prep 3: s[h][l] = 1 / max(||ptp[l] * w2[h]||_2, eps)  (pad cols are zero)
// ---------------------------------------------------------------------------
__global__ void invnorm_kernel(const float* __restrict__ ptp,
                               const float* __restrict__ w2,
                               float* __restrict__ s) {
    int i = blockIdx.x * blockDim.x + threadIdx.x;
    if (i >= NHEAD * L_DIM) return;
    int h = i / L_DIM, l = i % L_DIM;
    const float* row = ptp + (size_t)l * EPTP;
    const float* wr  = w2 + (size_t)h * EPTP;
    float acc = 0.0f;
    for (int e = 0; e < EPTP; ++e) {
        float v = row[e] * wr[e];
        acc += v * v;
    }
    s[i] = 1.0f / fmaxf(sqrtf(acc), 1e-12f);
}

// ---------------------------------------------------------------------------
// Scores: out[l][m] = mean_h sum_e hn_h[l][e]*hn_h[m][e], thresholded.
// Block = 16 rows x 128 cols (8 waves, one 16x16 tile each).
// PT rows of the tile and all w2 rows are staged in LDS via async b128.
// A frag (16x4 f32, ISA 7.12.2): lane 0-15 row M=lane K={0,1}; lane 16-31
// row M=lane-16 K={2,3}. B mirrors with N for M. C/D: VGPR r ->
// (M = r + 8*(lane>=16), N = lane&15).
// ---------------------------------------------------------------------------
__global__ void scores_kernel(const float* __restrict__ ptp,
                              const float* __restrict__ w2,
                              const float* __restrict__ s,
                              float* __restrict__ out) {
    __shared__ float ldsPT[16 * EPTP];     // 19456 B
    __shared__ float ldsW2[NHEAD * EPTP];  // 19456 B

    const int t       = threadIdx.x;
    const int lane    = t & 31;
    const int wave    = t >> 5;
    const int rowTile = blockIdx.x >> 4;   // 128 row tiles
    const int colGrp  = blockIdx.x & 15;   // 16 col groups of 128
    const int rowBase = rowTile << 4;
    const int colBase = (colGrp << 7) + (wave << 4);
    const int mn      = lane & 15;
    const int khalf   = (lane >> 4) << 1;  // 0 or 2

    // Stage 16 PT rows + 16 w2 rows (each 4864 floats = 1216 b128 chunks).
    {
        const unsigned ptBytes = (unsigned)((size_t)rowBase * EPTP * 4);
        for (int idx = t; idx < (16 * EPTP) / 4; idx += 256) {
            async_g2l_b128(lds_addr_of(&ldsPT[idx * 4]), ptBytes + idx * 16u, ptp);
            async_g2l_b128(lds_addr_of(&ldsW2[idx * 4]), idx * 16u, w2);
        }
        wait_async0();
    }
    __syncthreads();

    const float* rowB = ptp + (size_t)(colBase + mn) * EPTP;
    const float* ldsA = ldsPT + mn * EPTP;

    v8f c = {};
    for (int h = 0; h < NHEAD; ++h) {
        const float sA = s[h * L_DIM + rowBase + mn];
        const float sB = s[h * L_DIM + colBase + mn];
        const float* wr = ldsW2 + h * EPTP;
#pragma unroll 4
        for (int k = 0; k < EPTP; k += 4) {
            const int kk = k + khalf;
            v2f w  = *(const v2f*)(wr   + kk);
            v2f pa = *(const v2f*)(ldsA + kk);
            v2f pb = *(const v2f*)(rowB + kk);
            v2f a, b;
            a.x = pa.x * w.x * sA;  a.y = pa.y * w.y * sA;
            b.x = pb.x * w.x * sB;  b.y = pb.y * w.y * sB;
            c = __builtin_amdgcn_wmma_f32_16x16x4_f32(
                    false, a, false, b, (short)0, c, false, false);
        }
    }

    const int n     = colBase + mn;
    const int mBase = rowBase + ((lane >> 4) << 3);
#pragma unroll
    for (int r = 0; r < 8; ++r) {
        float v = c[r] * (1.0f / NHEAD);
        v = (v >= THRESH) ? v : -NEGINF;
        out[(size_t)(mBase + r) * L_DIM + n] = v;
    }
}

// ---------------------------------------------------------------------------
// Row softmax over L_DIM entries, in place. One 256-thread block per row.
// ---------------------------------------------------------------------------
__global__ void softmax_kernel(float* __restrict__ adj) {
    __shared__ float red[256];
    float* p = adj + (size_t)blockIdx.x * L_DIM;
    const int t = threadIdx.x;

    float m = -INFINITY;
    for (int j = t; j < L_DIM; j += 256) m = fmaxf(m, p[j]);
    red[t] = m; __syncthreads();
    for (int st = 128; st > 0; st >>= 1) {
        if (t < st) red[t] = fmaxf(red[t], red[t + st]);
        __syncthreads();
    }
    m = red[0]; __syncthreads();

    float sum = 0.0f;
    for (int j = t; j < L_DIM; j += 256) {
        float e = expf(p[j] - m);
        p[j] = e;
        sum += e;
    }
    red[t] = sum; __syncthreads();
    for (int st = 128; st > 0; st >>= 1) {
        if (t < st) red[t] += red[t + st];
        __syncthreads();
    }
    float inv = 1.0f / red[0];
    __syncthreads();
    for (int j = t; j < L_DIM; j += 256) p[j] *= inv;
}

// ---------------------------------------------------------------------------
// f32 WMMA GEMM: C = A[MxK] * B[KxN], N == 512, K % 64 == 0, fused relu.
// Block = one 16-row tile; 8 waves each own 64 cols (4 accumulators).
// A K-chunks (16x64 = 4KB) staged in LDS: one async b128 per thread.
// ---------------------------------------------------------------------------
__global__ void wmma_gemm_kernel(const float* __restrict__ A,
                                 const float* __restrict__ B,
                                 float* __restrict__ C,
                                 int M, int N, int K, int relu) {
    __shared__ float ldsA[16 * 64];

    const int t       = threadIdx.x;
    const int lane    = t & 31;
    const int wave    = t >> 5;
    const int rowBase = blockIdx.x << 4;
    const int colBase = wave << 6;          // 64 cols per wave
    const int mn      = lane & 15;
    const int khalf   = (lane >> 4) << 1;

    // per-thread stage coords: element = t*4 of the 16x64 chunk
    const int sRow = t >> 4;                // (t*4)/64
    const int sCol = (t & 15) << 2;         // (t*4)%64
    const unsigned ldsDst = lds_addr_of(&ldsA[t * 4]);

    v8f c0 = {}, c1 = {}, c2 = {}, c3 = {};
    const float* ldsRow = ldsA + mn * 64;

    for (int kc = 0; kc < K; kc += 64) {
        async_g2l_b128(ldsDst,
                       (unsigned)(((size_t)(rowBase + sRow) * K + kc + sCol) * 4),
                       A);
        wait_async0();
        __syncthreads();

        for (int k = 0; k < 64; k += 4) {
            const int kk = k + khalf;
            v2f a = *(const v2f*)(ldsRow + kk);
            const float* bp = B + (size_t)(kc + kk) * N + colBase + mn;
            v2f b0, b1, b2, b3;
            b0.x = bp[0];      b0.y = bp[N];
            b1.x = bp[16];     b1.y = bp[N + 16];
            b2.x = bp[32];     b2.y = bp[N + 32];
            b3.x = bp[48];     b3.y = bp[N + 48];
            c0 = __builtin_amdgcn_wmma_f32_16x16x4_f32(false, a, false, b0, (short)0, c0, false, false);
            c1 = __builtin_amdgcn_wmma_f32_16x16x4_f32(false, a, false, b1, (short)0, c1, false, false);
            c2 = __builtin_amdgcn_wmma_f32_16x16x4_f32(false, a, false, b2, (short)0, c2, false, false);
            c3 = __builtin_amdgcn_wmma_f32_16x16x4_f32(false, a, false, b3, (short)0, c3, false, false);
        }
        __syncthreads();   // protect LDS before next chunk overwrite
    }

    const int mBase = rowBase + ((lane >> 4) << 3);
    v8f acc[4] = {c0, c1, c2, c3};
#pragma unroll
    for (int sb = 0; sb < 4; ++sb) {
        const int n = colBase + (sb << 4) + mn;
#pragma unroll
        for (int r = 0; r < 8; ++r) {
            float v = acc[sb][r];
            if (relu) v = fmaxf(v, 0.0f);
            C[(size_t)(mBase + r) * N + n] = v;
        }
    }
}

// ---------------------------------------------------------------------------
// Launch
// ---------------------------------------------------------------------------
extern "C" void kernel_launch(void* const* d_in, const int* in_sizes, int n_in,
                              void* d_out, int out_size, void* d_ws, size_t ws_size,
                              hipStream_t stream) {
    const float* LE  = (const float*)d_in[0];  // [2048, 512]
    const float* PT  = (const float*)d_in[1];  // [2048, 300]
    const float* Wpt = (const float*)d_in[2];  // [16, 300]
    const float* W0  = (const float*)d_in[3];  // [512, 512]
    const float* W1  = (const float*)d_in[4];  // [512, 512]
    float* out = (float*)d_out;                // [2048, 512]

    float* ws  = (float*)d_ws;
    float* w2  = ws;                                  // 16*304   = 4,864
    float* s   = ws + 8192;                           // 16*2048  = 32,768
    float* ptp = ws + 49152;                          // 2048*304 = 622,592
    float* adj = ptp + (size_t)L_DIM * EPTP;          // 2048*2048
    float* t0  = adj + (size_t)L_DIM * L_DIM;         // 2048*512
    float* x1  = t0  + (size_t)L_DIM * E_DIM;         // 2048*512
    float* t1  = x1  + (size_t)L_DIM * E_DIM;         // 2048*512

    w2_kernel<<<(NHEAD * EPTP + 255) / 256, 256, 0, stream>>>(Wpt, w2);
    pad_pt_kernel<<<(L_DIM * EPTP + 255) / 256, 256, 0, stream>>>(PT, ptp);
    invnorm_kernel<<<(NHEAD * L_DIM + 255) / 256, 256, 0, stream>>>(ptp, w2, s);
    // cosine scores + threshold: 128 row tiles x 16 col groups
    scores_kernel<<<128 * 16, 256, 0, stream>>>(ptp, w2, s, adj);
    softmax_kernel<<<L_DIM, 256, 0, stream>>>(adj);
    // t0 = LE @ W0
    wmma_gemm_kernel<<<L_DIM / 16, 256, 0, stream>>>(LE, W0, t0, L_DIM, E_DIM, E_DIM, 0);
    // x1 = relu(adj @ t0)
    wmma_gemm_kernel<<<L_DIM / 16, 256, 0, stream>>>(adj, t0, x1, L_DIM, E_DIM, L_DIM, 1);
    // t1 = x1 @ W1
    wmma_gemm_kernel<<<L_DIM / 16, 256, 0, stream>>>(x1, W1, t1, L_DIM, E_DIM, E_DIM, 0);
    // out = relu(adj @ t1)
    wmma_gemm_kernel<<<L_DIM / 16, 256, 0, stream>>>(adj, t1, out, L_DIM, E_DIM, L_DIM, 1);
}